// Tile_Renderer_79413945303599
// MI455X (gfx1250) — compile-verified
//
#include <hip/hip_runtime.h>
#include <math.h>

// Problem constants (from reference): 48x48x3 image, (1,8,8,8,24) input.
#define NTRI   6144     // 512 cells * 6 faces * 2 tris
#define NGRP   1536     // groups of 4 triangles (one WMMA each)
#define NPIX   2304     // 48*48
#define NCHUNK 4        // triangle chunks (parallelism)
#define CHUNKG 384      // groups per chunk
#define IMG_W  48
#define IMG_H  48

typedef __attribute__((ext_vector_type(2))) float v2f;
typedef __attribute__((ext_vector_type(8))) float v8f;

struct XformParams {
  float view[16];           // fp32 view matrix (row-vector convention)
  float sx, sy, za, zb;     // ortho scalars: ndc = (cx*sx, cy*sy, cz*za+zb)
};

// Cube face/tri/vertex indices from the reference _cube_tris(); vertex i has
// coords (i&1, (i>>1)&1, (i>>2)&1).
__device__ const int FACE_V[36] = {
  0,1,3, 3,2,0,   4,5,7, 7,6,4,   1,5,4, 4,0,1,
  2,6,7, 7,3,2,   4,6,2, 2,0,4,   3,7,5, 5,1,3
};

// ---------------------------------------------------------------------------
// Kernel 1: triangle setup.  For each triangle, produce affine coefficients
// for (b0,b1,b2,depth) as functions of (px,py) and store them in the exact
// VGPR layout of the WMMA f32 16x16x4 A-matrix, 4 triangles per block:
//   blk[g*80 + m*4 + k]  : row m = 4*triLocal + func, k = {A(px),B(py),C,0}
//   blk[g*80 + 64 + j*4] : (r,g,b,0) face color of triangle j
// ---------------------------------------------------------------------------
__global__ __launch_bounds__(256)
void setup_kernel(const float* __restrict__ inp, float* __restrict__ blk,
                  XformParams P) {
  const int t = blockIdx.x * blockDim.x + threadIdx.x;
  if (t >= NTRI) return;

  const int cell = t / 12;
  const int face = (t % 12) >> 1;
  const int tri  = t & 1;

  const float* e = inp + cell * 24;
  const float fx = e[0], fy = e[1], fz = e[2];
  const float dx = e[3] - fx, dy = e[4] - fy, dz = e[5] - fz;
  const float cr = e[6 + 3 * face], cg = e[7 + 3 * face], cb = e[8 + 3 * face];
  const float gx = (float)(cell >> 6);
  const float gy = (float)((cell >> 3) & 7);
  const float gz = (float)(cell & 7);

  float xs[3], ys[3], zs[3];
#pragma unroll
  for (int k = 0; k < 3; ++k) {
    const int vi = FACE_V[face * 6 + tri * 3 + k];
    const float ux = (float)(vi & 1);
    const float uy = (float)((vi >> 1) & 1);
    const float uz = (float)((vi >> 2) & 1);
    const float x = ux * dx + fx + gx;            // _VCUBE*(to-frm)+frm + off
    const float y = uy * dy + fy + gy;
    const float z = uz * dz + fz + gz;
    const float wx = x - 25.0f;                   // v4 @ translation
    const float wy = y - 20.0f;
    const float wz = z - 25.0f;
    const float cx = wx * P.view[0] + wy * P.view[4] + wz * P.view[8]  + P.view[12];
    const float cy = wx * P.view[1] + wy * P.view[5] + wz * P.view[9]  + P.view[13];
    const float cz = wx * P.view[2] + wy * P.view[6] + wz * P.view[10] + P.view[14];
    const float nx = cx * P.sx;                   // ortho, clip.w == 1
    const float ny = cy * P.sy;
    const float nz = cz * P.za + P.zb;
    xs[k] = (nx * 0.5f + 0.5f) * (float)IMG_W;
    ys[k] = (0.5f - ny * 0.5f) * (float)IMG_H;
    zs[k] = nz;
  }

  const float x0 = xs[0], x1 = xs[1], x2 = xs[2];
  const float y0 = ys[0], y1 = ys[1], y2 = ys[2];
  const float area = (x1 - x0) * (y2 - y0) - (y1 - y0) * (x2 - x0);
  const bool valid = fabsf(area) > 1e-9f;

  float A0, B0, C0, A1, B1, C1, A2, B2, C2, Ad, Bd, Cd;
  if (valid) {
    // edge(a,b) = (bx-ax)*(PY-ay) - (by-ay)*(PX-ax), pre-divided by signed area
    A0 = -(y2 - y1) / area; B0 = (x2 - x1) / area;
    C0 = ((y2 - y1) * x1 - (x2 - x1) * y1) / area;
    A1 = -(y0 - y2) / area; B1 = (x0 - x2) / area;
    C1 = ((y0 - y2) * x2 - (x0 - x2) * y2) / area;
    A2 = -(y1 - y0) / area; B2 = (x1 - x0) / area;
    C2 = ((y1 - y0) * x0 - (x1 - x0) * y0) / area;
    Ad = A0 * zs[0] + A1 * zs[1] + A2 * zs[2];
    Bd = B0 * zs[0] + B1 * zs[1] + B2 * zs[2];
    Cd = C0 * zs[0] + C1 * zs[1] + C2 * zs[2];
  } else {
    // degenerate: barycentrics always -1 -> never "inside"
    A0 = B0 = A1 = B1 = A2 = B2 = 0.0f;
    C0 = C1 = C2 = -1.0f;
    Ad = Bd = Cd = 0.0f;
  }

  float* o = blk + (t >> 2) * 80;
  const int j = t & 3;
  const float rows[16] = {A0, B0, C0, 0.0f,  A1, B1, C1, 0.0f,
                          A2, B2, C2, 0.0f,  Ad, Bd, Cd, 0.0f};
#pragma unroll
  for (int q = 0; q < 16; ++q) o[j * 16 + q] = rows[q];
  o[64 + j * 4 + 0] = cr;
  o[64 + j * 4 + 1] = cg;
  o[64 + j * 4 + 2] = cb;
  o[64 + j * 4 + 3] = 0.0f;
}

// ---------------------------------------------------------------------------
// Kernel 2: WMMA raster.  One wave = 16 pixels x one triangle chunk.
// D = A(16 funcs x 4) @ B(4 x 16 pixels): per WMMA we get b0,b1,b2,depth of
// 4 triangles at 16 pixels.
//   - inside test: min3(b0,b1,b2) >= 0   (one v_min3 + one v_cmp)
//   - argmin:      track (bestD, bestG [cndmask w/ SGPR g], bestL [0/1]);
//                  absolute index materialized once after the loop.
// ---------------------------------------------------------------------------
__global__ __launch_bounds__(256)
void raster_kernel(const float* __restrict__ blk, float* __restrict__ partD,
                   int* __restrict__ partI) {
  const int lane  = threadIdx.x & 31;
  const int wave  = (blockIdx.x * blockDim.x + threadIdx.x) >> 5;
  const int chunk = wave & (NCHUNK - 1);
  const int tile  = wave >> 2;          // 144 pixel tiles
  const int lp = lane & 15;             // pixel slot (N column)
  const int hi = lane >> 4;             // half-wave: which K/func half
  const int p  = tile * 16 + lp;
  const float px = (float)(p % IMG_W) + 0.5f;
  const float py = (float)(p / IMG_W) + 0.5f;

  // B (4x16 f32): lanes 0-15 -> K0=px, K1=py ; lanes 16-31 -> K2=1, K3=0
  v2f bmat;
  bmat.x = hi ? 1.0f : px;
  bmat.y = hi ? 0.0f : py;

  float bestD = __builtin_inff();
  int   bestG = 1 << 27;                // sentinel group (no hit)
  int   bestL = 0;                      // local triangle bit within the pair

  const int aoff = lp * 2 + hi;         // v2f index into a group's A block
  const int gbeg = chunk * CHUNKG;

#pragma unroll 4
  for (int g = gbeg; g < gbeg + CHUNKG; ++g) {
    const v2f* ap = (const v2f*)(blk + g * 80);
    __builtin_prefetch((const void*)(blk + (g + 4) * 80), 0, 1);
    const v2f amat = ap[aoff];          // coalesced b64 load in A layout
    v8f acc = {0.f, 0.f, 0.f, 0.f, 0.f, 0.f, 0.f, 0.f};
    acc = __builtin_amdgcn_wmma_f32_16x16x4_f32(
        false, amat, false, bmat, (short)0, acc, false, false);

    // lane<16 sees funcs 0-7 (tris 4g+0,4g+1); lane>=16 sees 8-15 (4g+2,4g+3)
    {
      const float mn = fminf(fminf(acc[0], acc[1]), acc[2]);
      const float d  = acc[3];
      const bool take = (mn >= 0.f) && (d < bestD);  // strict < keeps earliest
      bestD = take ? d : bestD;
      bestG = take ? g : bestG;
      bestL = take ? 0 : bestL;
    }
    {
      const float mn = fminf(fminf(acc[4], acc[5]), acc[6]);
      const float d  = acc[7];
      const bool take = (mn >= 0.f) && (d < bestD);
      bestD = take ? d : bestD;
      bestG = take ? g : bestG;
      bestL = take ? 1 : bestL;
    }
  }

  // Materialize absolute triangle index once.
  int bestT = (bestG < (1 << 27)) ? (bestG * 4 + hi * 2 + bestL) : (1 << 30);

  // Fold the two half-wave partners (same pixel, disjoint triangle sets).
  const float oD = __shfl_xor(bestD, 16);
  const int   oT = __shfl_xor(bestT, 16);
  const bool take = (oD < bestD) || ((oD == bestD) && (oT < bestT));
  bestD = take ? oD : bestD;
  bestT = take ? oT : bestT;

  if (lane < 16) {
    partD[chunk * NPIX + p] = bestD;
    partI[chunk * NPIX + p] = bestT;
  }
}

// ---------------------------------------------------------------------------
// Kernel 3: merge chunks (argmin with first-index tie-break, matching
// jnp.argmin), recompute winning barycentrics, shade.
// ---------------------------------------------------------------------------
__global__ __launch_bounds__(256)
void merge_kernel(const float* __restrict__ blk, const float* __restrict__ partD,
                  const int* __restrict__ partI, float* __restrict__ out) {
  const int p = blockIdx.x * blockDim.x + threadIdx.x;
  if (p >= NPIX) return;
  float bd = __builtin_inff();
  int   bt = 1 << 30;
#pragma unroll
  for (int c = 0; c < NCHUNK; ++c) {
    const float d = partD[c * NPIX + p];
    const int  tt = partI[c * NPIX + p];
    const bool take = (d < bd) || ((d == bd) && (tt < bt));
    bd = take ? d : bd;
    bt = take ? tt : bt;
  }
  float r = 0.f, g = 0.f, b = 0.f;   // background is zeros
  if (bt < (1 << 30)) {
    const float px = (float)(p % IMG_W) + 0.5f;
    const float py = (float)(p / IMG_W) + 0.5f;
    const int grp = bt >> 2, j = bt & 3;
    const float* base = blk + grp * 80;
    const float g0 = base[j * 16 + 0] * px + base[j * 16 + 1] * py + base[j * 16 + 2];
    const float g1 = base[j * 16 + 4] * px + base[j * 16 + 5] * py + base[j * 16 + 6];
    const float g2 = base[j * 16 + 8] * px + base[j * 16 + 9] * py + base[j * 16 + 10];
    const float* col = base + 64 + j * 4;
    r = g0 * col[0] + g1 * col[0] + g2 * col[0];   // all 3 vertex colors equal
    g = g0 * col[1] + g1 * col[1] + g2 * col[1];
    b = g0 * col[2] + g1 * col[2] + g2 * col[2];
  }
  float* o = out + p * 3;
  o[0] = r; o[1] = g; o[2] = b;
}

// ---------------------------------------------------------------------------
// Host: build view matrix exactly like reference: ((I@R1)@R2)@R3 in fp32,
// each R from double-precision Rodrigues (then M[:3,:3] = R.T).
// ---------------------------------------------------------------------------
static void rodrigues_host(double r0, double r1, double r2, float M[16]) {
  const double th = sqrt(r0 * r0 + r1 * r1 + r2 * r2);
  double R[3][3] = {{1, 0, 0}, {0, 1, 0}, {0, 0, 1}};
  if (th >= 1e-12) {
    const double k0 = r0 / th, k1 = r1 / th, k2 = r2 / th;
    const double K[3][3] = {{0, -k2, k1}, {k2, 0, -k0}, {-k1, k0, 0}};
    double KK[3][3];
    for (int i = 0; i < 3; ++i)
      for (int j = 0; j < 3; ++j)
        KK[i][j] = K[i][0] * K[0][j] + K[i][1] * K[1][j] + K[i][2] * K[2][j];
    const double s = sin(th), c = 1.0 - cos(th);
    for (int i = 0; i < 3; ++i)
      for (int j = 0; j < 3; ++j)
        R[i][j] = (i == j ? 1.0 : 0.0) + s * K[i][j] + c * KK[i][j];
  }
  for (int i = 0; i < 16; ++i) M[i] = (i % 5 == 0) ? 1.0f : 0.0f;
  for (int i = 0; i < 3; ++i)
    for (int j = 0; j < 3; ++j) M[i * 4 + j] = (float)R[j][i];  // R.T
}

static void mm4f(const float* A, const float* B, float* C) {
  for (int i = 0; i < 4; ++i)
    for (int j = 0; j < 4; ++j) {
      float s = A[i * 4 + 0] * B[0 * 4 + j];
      s += A[i * 4 + 1] * B[1 * 4 + j];
      s += A[i * 4 + 2] * B[2 * 4 + j];
      s += A[i * 4 + 3] * B[3 * 4 + j];
      C[i * 4 + j] = s;
    }
}

extern "C" void kernel_launch(void* const* d_in, const int* in_sizes, int n_in,
                              void* d_out, int out_size, void* d_ws, size_t ws_size,
                              hipStream_t stream) {
  (void)in_sizes; (void)n_in; (void)out_size; (void)ws_size;
  const float* inp = (const float*)d_in[0];
  float* out = (float*)d_out;

  // ws layout: [ NGRP*80 floats setup blocks | NCHUNK*NPIX partD | partI ]
  float* blk   = (float*)d_ws;
  float* partD = (float*)((char*)d_ws + (size_t)NGRP * 80 * 4);
  int*   partI = (int*)((char*)d_ws + (size_t)NGRP * 80 * 4 +
                        (size_t)NCHUNK * NPIX * 4);

  XformParams xp;
  float R1[16], R2[16], R3[16], R12[16];
  rodrigues_host(-0.7853982, 0.0, 0.0, R1);
  rodrigues_host(0.0, 0.7853982, 0.0, R2);
  rodrigues_host(0.0, 0.0, 0.7853982, R3);
  mm4f(R1, R2, R12);
  mm4f(R12, R3, xp.view);
  xp.sx = (float)(2.0 / 100.0);
  xp.sy = (float)(2.0 / 100.0);               // aspect = 48/48 = 1
  xp.za = (float)(-1.0 / (1000.0 - 0.1));
  xp.zb = (float)(-0.1 / (1000.0 - 0.1));

  setup_kernel<<<(NTRI + 255) / 256, 256, 0, stream>>>(inp, blk, xp);
  raster_kernel<<<(NPIX / 16) * NCHUNK / 8, 256, 0, stream>>>(blk, partD, partI);
  merge_kernel<<<(NPIX + 255) / 256, 256, 0, stream>>>(blk, partD, partI, out);
}